// PhysioNetGeoLIF_4Class_58780922413198
// MI455X (gfx1250) — compile-verified
//
#include <hip/hip_runtime.h>
#include <hip/hip_bf16.h>

typedef float v2f __attribute__((ext_vector_type(2)));
typedef float v8f __attribute__((ext_vector_type(8)));

#define B_   256
#define T_   4096
#define C_   64
#define NC_  4
#define TDA_ 50
#define LEAK_ 0.9f

// ---------------------------------------------------------------------------
// Kernel 1: W_fused[m][c] = sum_n lateral[n][m] * W_spatial[n][c]   (4 x 64)
// ---------------------------------------------------------------------------
__global__ void fuse_w_kernel(const float* __restrict__ W_spatial,
                              const float* __restrict__ lateral,
                              float* __restrict__ wf) {
  int i = threadIdx.x;          // 256 threads: m = i/64, c = i%64
  int m = i >> 6, c = i & 63;
  float acc = 0.f;
#pragma unroll
  for (int n = 0; n < NC_; ++n)
    acc = fmaf(lateral[n * NC_ + m], W_spatial[n * C_ + c], acc);
  wf[m * C_ + c] = acc;
}

// ---------------------------------------------------------------------------
// Kernel 2: dynamic threshold th[b][n] = 1 + 0.3*sigmoid(tda[b]·W_tda[n]+b[n])
// ---------------------------------------------------------------------------
__global__ void thresh_kernel(const float* __restrict__ tda,
                              const float* __restrict__ W_tda,
                              const float* __restrict__ b_tda,
                              float* __restrict__ th) {
  int i = blockIdx.x * blockDim.x + threadIdx.x;  // 0..1023
  int b = i >> 2, n = i & 3;
  float acc = b_tda[n];
  const float* tb = tda + b * TDA_;
  const float* wn = W_tda + n * TDA_;
#pragma unroll 10
  for (int j = 0; j < TDA_; ++j) acc = fmaf(tb[j], wn[j], acc);
  float sg = 1.0f / (1.0f + __expf(-acc));
  th[i] = 1.0f + 0.3f * sg;
}

// ---------------------------------------------------------------------------
// Kernel 3: proj[g][n] = x[g][:] . W_fused[n][:]   via V_WMMA_F32_16X16X4_F32
//   Per-wave 16-row tiles, double-buffered in LDS via async-to-LDS loads.
//   Only loads ride ASYNCcnt in this kernel -> in-order completion, so a
//   counted wait (<=8) overlaps next-tile HBM fetch with current WMMA chain.
// ---------------------------------------------------------------------------
__global__ void __launch_bounds__(256)
proj_wmma_kernel(const float* __restrict__ x,
                 const float* __restrict__ wf,
                 float* __restrict__ proj) {
  __shared__ float wf_lds[NC_ * C_];
  __shared__ float atile[8 * 2 * 16 * 68];   // 8 waves x 2 bufs x (16x64 pad68)

  const int tid  = threadIdx.x;
  const int lane = tid & 31;
  const int wave = tid >> 5;

  if (tid < NC_ * C_) wf_lds[tid] = wf[tid];
  __syncthreads();                 // wf_lds is the only cross-wave LDS data

  const int n    = lane & 15;
  const int m    = lane & 15;
  const int koff = (lane < 16) ? 0 : 2;

  // B fragments (constant across all tiles): N<4 columns hold W_fused.
  v2f bfrag[16];
#pragma unroll
  for (int ks = 0; ks < 16; ++ks) {
    float bx = 0.f, by = 0.f;
    if (n < NC_) {
      bx = wf_lds[n * C_ + ks * 4 + koff];
      by = wf_lds[n * C_ + ks * 4 + koff + 1];
    }
    bfrag[ks].x = bx;
    bfrag[ks].y = by;
  }

  float* buf0 = atile + wave * (2 * 16 * 68);
  float* buf1 = buf0 + 16 * 68;

  auto issue_tile_loads = [&](int tile, float* buf) {
    const float* src = x + (size_t)tile * 16 * C_;
#pragma unroll
    for (int j = 0; j < 8; ++j) {
      int f  = lane + j * 32;            // 0..255 float4s of the 16x64 tile
      int r  = f >> 4, c4 = f & 15;
      unsigned           ldsb = (unsigned)(uintptr_t)(buf + r * 68 + c4 * 4);
      unsigned long long ga   = (unsigned long long)(uintptr_t)(src + f * 4);
      asm volatile("global_load_async_to_lds_b128 %0, %1, off"
                   :: "v"(ldsb), "v"(ga) : "memory");
    }
  };

  issue_tile_loads(blockIdx.x * 64 + wave, buf0);

#pragma unroll 1
  for (int it = 0; it < 8; ++it) {
    float* cur = (it & 1) ? buf1 : buf0;
    float* nxt = (it & 1) ? buf0 : buf1;
    if (it < 7) {
      issue_tile_loads(blockIdx.x * 64 + (it + 1) * 8 + wave, nxt);
      asm volatile("s_wait_asynccnt 0x8" ::: "memory");  // current tile landed
    } else {
      asm volatile("s_wait_asynccnt 0x0" ::: "memory");
    }

    // Preload all 16 A fragments, then run the WMMA chain back-to-back.
    v2f afrag[16];
#pragma unroll
    for (int ks = 0; ks < 16; ++ks) {
      afrag[ks].x = cur[m * 68 + ks * 4 + koff];
      afrag[ks].y = cur[m * 68 + ks * 4 + koff + 1];
    }

    v8f acc = {};
#pragma unroll
    for (int ks = 0; ks < 16; ++ks)
      acc = __builtin_amdgcn_wmma_f32_16x16x4_f32(
          false, afrag[ks], false, bfrag[ks], (short)0, acc, false, false);

    const size_t rowBase = (size_t)(blockIdx.x * 64 + it * 8 + wave) * 16;
    if (n < NC_) {
      const size_t mBase = rowBase + ((lane < 16) ? 0 : 8);
#pragma unroll
      for (int r = 0; r < 8; ++r)
        proj[(mBase + r) * NC_ + n] = acc[r];
    }
  }
}

// ---------------------------------------------------------------------------
// Kernel 4: serial LIF scan over T, chunked through LDS.
//   Block = 64 batches x 4 channels = 256 threads; each 4-thread b-group is
//   contained in one wave, so all LDS hand-offs need only wave-scalar waits.
//   Loads: global_load_async_to_lds_b128 ; stores: global_store_async_from_lds.
// ---------------------------------------------------------------------------
__global__ void __launch_bounds__(256)
lif_scan_kernel(const float* __restrict__ proj,
                const float* __restrict__ th,
                float* __restrict__ out) {
  __shared__ float sbuf[64 * 260];       // 64 b x (64 t x 4 n), pad 4 floats

  const int tid   = threadIdx.x;
  const int bBase = blockIdx.x * 64;
  const int bl = tid >> 2;               // local batch 0..63
  const int q  = tid & 3;                // lane-in-group for load/store phases
  const int n  = tid & 3;                // channel for scan

  const float thr = th[(bBase + bl) * NC_ + n];
  float mem = 0.f;

  const float* gsrc = proj + (size_t)(bBase + bl) * T_ * NC_;
  float*       gdst = out  + (size_t)(bBase + bl) * T_ * NC_;
  float*       lrow = sbuf + bl * 260;

#pragma unroll 1
  for (int chunk = 0; chunk < 64; ++chunk) {
    const int t0 = chunk * 64;

    // Previous chunk's async stores must finish before LDS is overwritten.
    asm volatile("s_wait_asynccnt 0x0" ::: "memory");

    // 16 async b128 loads per thread: float4 j == timestep t0+j of this batch.
#pragma unroll
    for (int jj = 0; jj < 16; ++jj) {
      int j = q + jj * 4;
      unsigned           ldsb = (unsigned)(uintptr_t)(lrow + j * 4);
      unsigned long long ga =
          (unsigned long long)(uintptr_t)(gsrc + (size_t)t0 * NC_ + j * 4);
      asm volatile("global_load_async_to_lds_b128 %0, %1, off"
                   :: "v"(ldsb), "v"(ga) : "memory");
    }
    if (chunk < 63)
      __builtin_prefetch(gsrc + (size_t)(t0 + 64) * NC_ + q * 64, 0, 0);
    asm volatile("s_wait_asynccnt 0x0" ::: "memory");

    // Serial scan of 64 steps from LDS (pad-260 -> conflict-free).
    float* p = lrow + n;
#pragma unroll 8
    for (int tl = 0; tl < 64; ++tl) {
      float v = p[tl * 4];
      mem = fmaf(LEAK_, mem, v);
      bool fire = (mem >= thr);
      p[tl * 4] = fire ? 1.0f : 0.0f;   // spikes overwrite inputs in place
      mem = fire ? 0.0f : mem;
    }
    asm volatile("s_wait_dscnt 0x0" ::: "memory");  // spikes visible in LDS

    // Async store spikes LDS -> global (coalesced float4 per timestep).
#pragma unroll
    for (int jj = 0; jj < 16; ++jj) {
      int j = q + jj * 4;
      unsigned           ldsb = (unsigned)(uintptr_t)(lrow + j * 4);
      unsigned long long ga =
          (unsigned long long)(uintptr_t)(gdst + (size_t)t0 * NC_ + j * 4);
      asm volatile("global_store_async_from_lds_b128 %0, %1, off"
                   :: "v"(ga), "v"(ldsb) : "memory");
    }
  }
  asm volatile("s_wait_asynccnt 0x0" ::: "memory");
}

// ---------------------------------------------------------------------------
extern "C" void kernel_launch(void* const* d_in, const int* in_sizes, int n_in,
                              void* d_out, int out_size, void* d_ws, size_t ws_size,
                              hipStream_t stream) {
  const float* x    = (const float*)d_in[0];   // [B,T,C]
  const float* tda  = (const float*)d_in[1];   // [B,TDA]
  const float* Wsp  = (const float*)d_in[2];   // [NC,C]
  const float* lat  = (const float*)d_in[3];   // [NC,NC]
  const float* Wt   = (const float*)d_in[4];   // [NC,TDA]
  const float* bt   = (const float*)d_in[5];   // [NC]
  float* out = (float*)d_out;                  // [B,T,NC] fp32
  float* ws  = (float*)d_ws;

  const size_t projElems = (size_t)B_ * T_ * NC_;   // 16 MB
  float* proj = ws;
  float* wfus = ws + projElems;
  float* thr  = wfus + NC_ * C_;

  fuse_w_kernel<<<1, 256, 0, stream>>>(Wsp, lat, wfus);
  thresh_kernel<<<4, 256, 0, stream>>>(tda, Wt, bt, thr);
  proj_wmma_kernel<<<1024, 256, 0, stream>>>(x, wfus, proj);
  lif_scan_kernel<<<4, 256, 0, stream>>>(proj, thr, out);
}